// AttentionDecoderRNN_24653112279082
// MI455X (gfx1250) — compile-verified
//
#include <hip/hip_runtime.h>
#include <hip/hip_bf16.h>
#include <math.h>

// ---------------------------------------------------------------------------
// AttentionDecoderRNN single step, MI455X (gfx1250), fp32 end-to-end.
// GEMVs via V_WMMA_F32_16X16X4_F32 (f32 A/B, f32 accum):
//   A(16x4): row 0 = activation chunk (lanes n==0 per half-wave), rest 0
//   B(4x16): weight tile; D row 0 (c[0], lanes 0..15) = 16 dot products.
// Inner loops are branch-free (lane-mask multiply instead of EXEC divergence)
// and K-unrolled x4 with loads hoisted ahead of the WMMA chain.
// All reductions fixed-order (LDS trees / shfl) -> bitwise deterministic.
// ---------------------------------------------------------------------------

typedef float v2f __attribute__((ext_vector_type(2)));
typedef float v8f __attribute__((ext_vector_type(8)));

#define H 1024
#define L 4096
#define VOCAB 29

#define WMMA_F32(a, b, c) \
    __builtin_amdgcn_wmma_f32_16x16x4_f32(false, (a), false, (b), (short)0, (c), false, false)

// y[rowBase + n] (+)= dot(W[rowBase+n, :K], x) + bias[rowBase+n]
// W row-major (rows x K). One wave per block; blockIdx.x = row tile.
__global__ __launch_bounds__(32)
void gemvT_wmma(const float* __restrict__ W, const float* __restrict__ x,
                const float* __restrict__ bias, float* __restrict__ y,
                int K, int rows, int accumulate, int relu)
{
    const int lane    = threadIdx.x;      // 0..31
    const int n       = lane & 15;        // B-tile column / output row offset
    const int khalf   = lane >> 4;        // 0: K={0,1}, 1: K={2,3}
    const int rowBase = blockIdx.x * 16;
    const float xmask = (n == 0) ? 1.0f : 0.0f;   // A row-0-only lane mask

    int r = rowBase + n;
    if (r >= rows) r = rows - 1;          // clamp loads; writes are guarded
    const float* __restrict__ Wrow = W + (size_t)r * (size_t)K;

    v8f c = {};
    for (int kb = 0; kb < K; kb += 16) {  // K always a multiple of 16 here
        const float* __restrict__ wp = Wrow + kb + 2 * khalf;
        const float* __restrict__ xp = x    + kb + 2 * khalf;
        __builtin_prefetch(wp + 256, 0, 1);         // global_prefetch_b8, 1KB ahead

        v2f b0, b1, b2, b3;
        b0.x = wp[0];  b0.y = wp[1];
        b1.x = wp[4];  b1.y = wp[5];
        b2.x = wp[8];  b2.y = wp[9];
        b3.x = wp[12]; b3.y = wp[13];

        v2f a0, a1, a2, a3;                          // branch-free A fragments
        a0.x = xp[0]  * xmask; a0.y = xp[1]  * xmask;
        a1.x = xp[4]  * xmask; a1.y = xp[5]  * xmask;
        a2.x = xp[8]  * xmask; a2.y = xp[9]  * xmask;
        a3.x = xp[12] * xmask; a3.y = xp[13] * xmask;

        c = WMMA_F32(a0, b0, c);
        c = WMMA_F32(a1, b1, c);
        c = WMMA_F32(a2, b2, c);
        c = WMMA_F32(a3, b3, c);
    }
    // D row M=0 lives in c[0], lanes 0..15 (N = lane).
    if (lane < 16) {
        int row = rowBase + lane;
        if (row < rows) {
            float v = c[0] + bias[row];
            if (accumulate) v += y[row];
            if (relu) v = v > 0.0f ? v : 0.0f;
            y[row] = v;
        }
    }
}

// ypart[blockIdx.y*N + colBase+n] = sum_{k in chunk} x[k] * E[k][colBase+n]
// E row-major (K x N); coalesced 16-wide row-segment loads.
__global__ __launch_bounds__(32)
void gemvE_wmma(const float* __restrict__ E, const float* __restrict__ x,
                float* __restrict__ ypart, int N, int Kchunk)
{
    const int lane    = threadIdx.x;
    const int n       = lane & 15;
    const int khalf   = lane >> 4;
    const int colBase = blockIdx.x * 16;
    const int kStart  = blockIdx.y * Kchunk;
    const float xmask = (n == 0) ? 1.0f : 0.0f;

    const float* __restrict__ Ebase = E + (size_t)kStart * (size_t)N + colBase + n;

    v8f c = {};
    for (int kb = 0; kb < Kchunk; kb += 16) {
        const float* __restrict__ ep = Ebase + (size_t)(kb + 2 * khalf) * (size_t)N;
        const float* __restrict__ xp = x + kStart + kb + 2 * khalf;
        __builtin_prefetch(ep + 16 * N, 0, 1);

        v2f b0, b1, b2, b3;
        b0.x = ep[0];              b0.y = ep[(size_t)N];
        b1.x = ep[(size_t)4 * N];  b1.y = ep[(size_t)5 * N];
        b2.x = ep[(size_t)8 * N];  b2.y = ep[(size_t)9 * N];
        b3.x = ep[(size_t)12 * N]; b3.y = ep[(size_t)13 * N];

        v2f a0, a1, a2, a3;
        a0.x = xp[0]  * xmask; a0.y = xp[1]  * xmask;
        a1.x = xp[4]  * xmask; a1.y = xp[5]  * xmask;
        a2.x = xp[8]  * xmask; a2.y = xp[9]  * xmask;
        a3.x = xp[12] * xmask; a3.y = xp[13] * xmask;

        c = WMMA_F32(a0, b0, c);
        c = WMMA_F32(a1, b1, c);
        c = WMMA_F32(a2, b2, c);
        c = WMMA_F32(a3, b3, c);
    }
    if (lane < 16)
        ypart[(size_t)blockIdx.y * (size_t)N + colBase + lane] = c[0];
}

// cat1 = [ emb[tok] | h0 ]  (2048)
__global__ __launch_bounds__(256)
void build_cat1_k(const float* __restrict__ emb, const int* __restrict__ tok,
                  const float* __restrict__ h0, float* __restrict__ cat1)
{
    int i = blockIdx.x * 256 + threadIdx.x;
    if (i < H)          cat1[i] = emb[(size_t)tok[0] * H + i];
    else if (i < 2 * H) cat1[i] = h0[i - H];
}

// softmax over 4096 logits; writes to ws copy and to d_out attn_weights slot.
__global__ __launch_bounds__(1024)
void softmax4096_k(const float* __restrict__ logits,
                   float* __restrict__ w_ws, float* __restrict__ w_out)
{
    __shared__ float red[1024];
    int t = threadIdx.x;
    float v0 = logits[t], v1 = logits[t + 1024],
          v2 = logits[t + 2048], v3 = logits[t + 3072];

    float m = fmaxf(fmaxf(v0, v1), fmaxf(v2, v3));
    red[t] = m; __syncthreads();
    for (int s = 512; s > 0; s >>= 1) {
        if (t < s) red[t] = fmaxf(red[t], red[t + s]);
        __syncthreads();
    }
    float M = red[0]; __syncthreads();

    float e0 = expf(v0 - M), e1 = expf(v1 - M),
          e2 = expf(v2 - M), e3 = expf(v3 - M);
    red[t] = (e0 + e1) + (e2 + e3); __syncthreads();
    for (int s = 512; s > 0; s >>= 1) {
        if (t < s) red[t] += red[t + s];
        __syncthreads();
    }
    float inv = 1.0f / red[0];

    float w0 = e0 * inv, w1 = e1 * inv, w2 = e2 * inv, w3 = e3 * inv;
    w_ws[t] = w0;  w_ws[t + 1024] = w1;  w_ws[t + 2048] = w2;  w_ws[t + 3072] = w3;
    w_out[t] = w0; w_out[t + 1024] = w1; w_out[t + 2048] = w2; w_out[t + 3072] = w3;
}

// cat2 = [ cat1[0:1024] (embedded) | sum of 4 attn_applied partials ]
__global__ __launch_bounds__(256)
void reduce_cat2_k(const float* __restrict__ parts, const float* __restrict__ cat1,
                   float* __restrict__ cat2)
{
    int i = blockIdx.x * 256 + threadIdx.x;
    if (i < H) {
        cat2[i] = cat1[i];
    } else if (i < 2 * H) {
        int j = i - H;
        cat2[i] = (parts[j] + parts[H + j]) + (parts[2 * H + j] + parts[3 * H + j]);
    }
}

// LSTM cell, PyTorch gate order i,f,g,o packed in gates[4096].
__global__ __launch_bounds__(256)
void lstm_cell_k(const float* __restrict__ gates, const float* __restrict__ c0,
                 float* __restrict__ h_ws, float* __restrict__ h_out,
                 float* __restrict__ c_out)
{
    int t = blockIdx.x * 256 + threadIdx.x;   // 0..1023
    float ig = gates[t], fg = gates[H + t], gg = gates[2 * H + t], og = gates[3 * H + t];
    float si = 1.0f / (1.0f + expf(-ig));
    float sf = 1.0f / (1.0f + expf(-fg));
    float so = 1.0f / (1.0f + expf(-og));
    float tg = tanhf(gg);
    float cn = sf * c0[t] + si * tg;
    float hn = so * tanhf(cn);
    h_ws[t] = hn; h_out[t] = hn; c_out[t] = cn;
}

// logits29[r] = dot(out_W[r,:], h) + out_b[r]
__global__ __launch_bounds__(256)
void out_logits_k(const float* __restrict__ W, const float* __restrict__ h,
                  const float* __restrict__ b, float* __restrict__ y)
{
    __shared__ float red[256];
    int r = blockIdx.x, t = threadIdx.x;
    float acc = 0.0f;
    for (int j = t; j < H; j += 256) acc += W[(size_t)r * H + j] * h[j];
    red[t] = acc; __syncthreads();
    for (int s = 128; s > 0; s >>= 1) {
        if (t < s) red[t] += red[t + s];
        __syncthreads();
    }
    if (t == 0) y[r] = red[0] + b[r];
}

// log_softmax over 29 logits with one wave32 (shfl reductions).
__global__ __launch_bounds__(32)
void logsoftmax29_k(const float* __restrict__ lg, float* __restrict__ out)
{
    int l = threadIdx.x;
    float v = (l < VOCAB) ? lg[l] : -3.402823466e38f;
    float m = v;
    for (int off = 16; off > 0; off >>= 1) m = fmaxf(m, __shfl_xor(m, off));
    float e = (l < VOCAB) ? expf(v - m) : 0.0f;
    float s = e;
    for (int off = 16; off > 0; off >>= 1) s += __shfl_xor(s, off);
    float lse = logf(s);
    if (l < VOCAB) out[l] = v - m - lse;
}

extern "C" void kernel_launch(void* const* d_in, const int* in_sizes, int n_in,
                              void* d_out, int out_size, void* d_ws, size_t ws_size,
                              hipStream_t stream)
{
    (void)in_sizes; (void)n_in; (void)out_size; (void)ws_size;
    const int*   tok   = (const int*)  d_in[0];
    const float* h0    = (const float*)d_in[1];   // (1,1,H)
    const float* c0    = (const float*)d_in[2];   // (1,1,H)
    const float* enc   = (const float*)d_in[3];   // (L,H)
    const float* emb   = (const float*)d_in[4];   // (V,H)
    const float* attnW = (const float*)d_in[5];   // (L,2H)
    const float* attnb = (const float*)d_in[6];   // (L,)
    const float* combW = (const float*)d_in[7];   // (H,2H)
    const float* combb = (const float*)d_in[8];   // (H,)
    const float* Wih   = (const float*)d_in[9];   // (4H,H)
    const float* Whh   = (const float*)d_in[10];  // (4H,H)
    const float* bih   = (const float*)d_in[11];  // (4H,)
    const float* bhh   = (const float*)d_in[12];  // (4H,)
    const float* outW  = (const float*)d_in[13];  // (V,H)
    const float* outb  = (const float*)d_in[14];  // (V,)

    float* out      = (float*)d_out;
    float* out_logp = out;                  // 29
    float* out_h    = out + VOCAB;          // 1024
    float* out_c    = out + VOCAB + H;      // 1024
    float* out_aw   = out + VOCAB + 2 * H;  // 4096

    float* ws    = (float*)d_ws;
    float* cat1  = ws;           // 2048
    float* alog  = ws + 2048;    // 4096  attention logits
    float* aw    = ws + 6144;    // 4096  attention weights (ws copy)
    float* parts = ws + 10240;   // 4096  attn_applied partials (4 x 1024)
    float* cat2  = ws + 14336;   // 2048  [embedded | attn_applied]
    float* xrel  = ws + 16384;   // 1024  relu(combine)
    float* gates = ws + 17408;   // 4096
    float* hnew  = ws + 21504;   // 1024
    float* lg29  = ws + 22528;   // 32

    // 1) cat1 = [emb[tok] | h0]
    build_cat1_k<<<8, 256, 0, stream>>>(emb, tok, h0, cat1);
    // 2) attention logits = cat1 @ attn_W.T + attn_b          (4096 rows, K=2048)
    gemvT_wmma<<<L / 16, 32, 0, stream>>>(attnW, cat1, attnb, alog, 2 * H, L, 0, 0);
    // 3) softmax over L
    softmax4096_k<<<1, 1024, 0, stream>>>(alog, aw, out_aw);
    // 4) attn_applied = attn_w @ enc  (K-split x4, coalesced)
    gemvE_wmma<<<dim3(H / 16, 4), 32, 0, stream>>>(enc, aw, parts, H, L / 4);
    // 5) cat2 = [embedded | attn_applied]
    reduce_cat2_k<<<8, 256, 0, stream>>>(parts, cat1, cat2);
    // 6) x = relu(cat2 @ comb_W.T + comb_b)                   (1024 rows, K=2048)
    gemvT_wmma<<<H / 16, 32, 0, stream>>>(combW, cat2, combb, xrel, 2 * H, H, 0, 1);
    // 7) gates = x @ W_ih.T + b_ih + h0 @ W_hh.T + b_hh       (4096 rows, K=1024)
    gemvT_wmma<<<(4 * H) / 16, 32, 0, stream>>>(Wih, xrel, bih, gates, H, 4 * H, 0, 0);
    gemvT_wmma<<<(4 * H) / 16, 32, 0, stream>>>(Whh, h0, bhh, gates, H, 4 * H, 1, 0);
    // 8) LSTM cell -> h_new, c_new
    lstm_cell_k<<<4, 256, 0, stream>>>(gates, c0, hnew, out_h, out_c);
    // 9) output head + log_softmax
    out_logits_k<<<VOCAB, 256, 0, stream>>>(outW, hnew, outb, lg29);
    logsoftmax29_k<<<1, 32, 0, stream>>>(lg29, out_logp);
}